// ASReader_31765578121391
// MI455X (gfx1250) — compile-verified
//
#include <hip/hip_runtime.h>
#include <math.h>

// Problem constants (match reference)
constexpr int B_  = 32;
constexpr int T_  = 4096;
constexpr int TQ_ = 64;
constexpr int D_  = 256;
constexpr int S_  = 64;

constexpr int CHUNK = 64;   // K-floats staged per LDS buffer
constexpr int PADF  = 68;   // padded LDS row stride (floats); 68 % 64 == 4 ->
                            // A-fragment ds_load_b64 hits all 64 banks once

typedef float v2f __attribute__((ext_vector_type(2)));
typedef float v8f __attribute__((ext_vector_type(8)));

// GCC-style int4 vector to match the async-builtin's expected pointee type
typedef int vi4 __attribute__((vector_size(16)));
typedef __attribute__((address_space(1))) vi4 gvi4;   // global int4
typedef __attribute__((address_space(3))) vi4 lvi4;   // LDS int4

// ---------------------------------------------------------------------------
// CDNA5 async global->LDS copy (ASYNCcnt path). Coalesced: lanes 0..15 of a
// half cover one contiguous 256B row segment; 512B per instruction.
// ---------------------------------------------------------------------------
__device__ __forceinline__ void async_copy_b128(const float* g, float* l) {
#if __has_builtin(__builtin_amdgcn_global_load_async_to_lds_b128)
  __builtin_amdgcn_global_load_async_to_lds_b128(
      (gvi4*)(size_t)g, (lvi4*)l, /*offset=*/0, /*cpol=*/0);
#else
  unsigned loff = (unsigned)(size_t)(__attribute__((address_space(3))) float*)l;
  asm volatile("global_load_async_to_lds_b128 %0, %1, off"
               :: "v"(loff), "v"(g)
               : "memory");
#endif
}

__device__ __forceinline__ void wait_async_le8() {
  asm volatile("s_wait_asynccnt 0x8" ::: "memory");
}
__device__ __forceinline__ void wait_async_0() {
  asm volatile("s_wait_asynccnt 0x0" ::: "memory");
}

// ---------------------------------------------------------------------------
// Kernel 1: query max-pool over TQ.  qpool[b,d] = max_t query[b,t,d]
// ---------------------------------------------------------------------------
__global__ void asr_qpool_kernel(const float* __restrict__ q,
                                 float* __restrict__ qpool) {
  const int b = blockIdx.x;
  const int d = threadIdx.x;  // blockDim.x == 256 == D
  const float* p = q + (size_t)b * TQ_ * D_ + d;
  float m = -INFINITY;
#pragma unroll 8
  for (int t = 0; t < TQ_; ++t) m = fmaxf(m, p[(size_t)t * D_]);
  qpool[b * D_ + d] = m;
}

// ---------------------------------------------------------------------------
// Kernel 2: scores[b,t] = doc[b,t,:] . qpool[b,:]  via V_WMMA_F32_16X16X4_F32
//   - one wave per 16-token tile; doc tile streamed through LDS with
//     double-buffered global_load_async_to_lds_b128 (ASYNCcnt split waits)
//   - A 16x4 f32 fragment read from padded LDS (lane l: row l%16,
//     K = 2*(l>>4)+v) -> conflict-free ds_load_b64
//   - B = query chunk replicated across all 16 N columns (LDS broadcast)
//   - C/D: token m lives in VGPR m%8 of lane 0 (m<8) / lane 16 (m>=8)
// ---------------------------------------------------------------------------
__global__ void __launch_bounds__(256)
asr_scores_wmma_kernel(const float* __restrict__ doc,
                       const float* __restrict__ qpool,
                       float* __restrict__ scores) {
  __shared__ float qs[D_];
  __shared__ float tilebuf[8][2][16 * PADF];  // per-wave double buffer

  // 256 tiles per batch, 8 waves per block -> 32 blocks per batch
  const int b    = blockIdx.x >> 5;
  const int wave = threadIdx.x >> 5;
  const int lane = threadIdx.x & 31;
  const int tile = ((blockIdx.x & 31) << 3) + wave;
  const int tokBase = tile << 4;
  const int half = lane >> 4;   // 0/1
  const int mrow = lane & 15;
  const int col4 = mrow << 2;   // coalesced-copy column (floats)

  const float* tileG = doc + ((size_t)b * T_ + tokBase) * D_;

  // stage qpool row for this batch
  qs[threadIdx.x] = qpool[b * D_ + threadIdx.x];

  // prologue: issue chunk 0 (8 async b128 -> 16 rows x 64 floats)
  {
    float* lb = &tilebuf[wave][0][0];
#pragma unroll
    for (int j = 0; j < 8; ++j) {
      const int r = j * 2 + half;
      async_copy_b128(tileG + (size_t)r * D_ + col4, lb + r * PADF + col4);
    }
  }
  __syncthreads();

  v8f c = {0.f, 0.f, 0.f, 0.f, 0.f, 0.f, 0.f, 0.f};

#pragma unroll
  for (int ch = 0; ch < 4; ++ch) {
    if (ch < 3) {
      // issue chunk ch+1 into the other buffer, then wait for chunk ch
      float* lb = &tilebuf[wave][(ch + 1) & 1][0];
      const float* gsrc = tileG + (ch + 1) * CHUNK;
#pragma unroll
      for (int j = 0; j < 8; ++j) {
        const int r = j * 2 + half;
        async_copy_b128(gsrc + (size_t)r * D_ + col4, lb + r * PADF + col4);
      }
      wait_async_le8();   // in-order done => chunk ch has landed
    } else {
      wait_async_0();     // last chunk
    }

    const float* aRow = &tilebuf[wave][ch & 1][0] + mrow * PADF + 2 * half;
    const float* bRow = &qs[ch * CHUNK + 2 * half];
#pragma unroll
    for (int kk = 0; kk < CHUNK; kk += 4) {
      v2f a  = *(const v2f*)(aRow + kk);   // ds_load_b64, bank-conflict-free
      v2f bq = *(const v2f*)(bRow + kk);   // ds_load_b64 broadcast
      c = __builtin_amdgcn_wmma_f32_16x16x4_f32(
          /*neg_a=*/false, a, /*neg_b=*/false, bq,
          /*c_mod=*/(short)0, c, /*reuse_a=*/false, /*reuse_b=*/false);
    }
  }

  float* sOut = scores + (size_t)b * T_ + tokBase;
  if (lane == 0) {
#pragma unroll
    for (int r = 0; r < 8; ++r) sOut[r] = c[r];
  } else if (lane == 16) {
#pragma unroll
    for (int r = 0; r < 8; ++r) sOut[8 + r] = c[r];
  }
}

// ---------------------------------------------------------------------------
// Kernel 3: masked softmax (faithful softmax_mask) + span rescale + renorm.
// One block (256 threads) per batch row; d_out used as probs scratch.
// ---------------------------------------------------------------------------
__global__ void asr_finalize_kernel(const float* __restrict__ scores,
                                    const float* __restrict__ mask,
                                    const int*   __restrict__ spans,
                                    const float* __restrict__ sscore,
                                    float* __restrict__ out) {
  const int b   = blockIdx.x;
  const int tid = threadIdx.x;  // 0..255

  __shared__ float red[256];
  __shared__ float sMax, sLogDen, sTot;
  __shared__ int   spStart[S_], spEnd[S_];
  __shared__ float spScale[S_];

  const float* sc = scores + (size_t)b * T_;
  const float* mk = mask   + (size_t)b * T_;
  float*       po = out    + (size_t)b * T_;

  // --- max of (scores * mask) ---
  float lmax = -INFINITY;
  for (int t = tid; t < T_; t += 256) lmax = fmaxf(lmax, sc[t] * mk[t]);
  red[tid] = lmax;
  __syncthreads();
  for (int s = 128; s > 0; s >>= 1) {
    if (tid < s) red[tid] = fmaxf(red[tid], red[tid + s]);
    __syncthreads();
  }
  if (tid == 0) sMax = red[0];
  __syncthreads();
  const float mx = sMax;

  // --- log denominator ---
  float lsum = 0.f;
  for (int t = tid; t < T_; t += 256) {
    float m_ = mk[t];
    float sh = (sc[t] * m_ - mx) * m_;
    lsum += __expf(sh) * m_;
  }
  __syncthreads();
  red[tid] = lsum;
  __syncthreads();
  for (int s = 128; s > 0; s >>= 1) {
    if (tid < s) red[tid] += red[tid + s];
    __syncthreads();
  }
  if (tid == 0) sLogDen = logf(red[0]);
  __syncthreads();
  const float logden = sLogDen;

  // --- probs = exp((shifted - logden) * mask) * mask -> stash in out ---
  for (int t = tid; t < T_; t += 256) {
    float m_ = mk[t];
    float sh = (sc[t] * m_ - mx) * m_;
    float lp = (sh - logden) * m_;
    po[t] = __expf(lp) * m_;
  }
  __syncthreads();

  // --- per-span mean and scale (generic from the spans input) ---
  if (tid < S_) {
    int st = spans[((size_t)b * S_ + tid) * 2 + 0];
    int en = spans[((size_t)b * S_ + tid) * 2 + 1];
    int lo = st > 0 ? st : 0;
    int hi = en < T_ ? en : T_;
    int count = hi - lo;
    if (count < 0) count = 0;
    float ssum = 0.f;
    for (int t = lo; t < hi; ++t) ssum += po[t];
    float mean  = ssum / (float)(count > 1 ? count : 1);
    float scale = (mean != 0.0f) ? (sscore[(size_t)b * S_ + tid] / mean) : 1.0f;
    spStart[tid] = st;
    spEnd[tid]   = en;
    spScale[tid] = scale;
  }
  __syncthreads();

  // --- token_scale = sum_s memb*scale + (1 - covered); apply and sum ---
  float ltot = 0.f;
  for (int t = tid; t < T_; t += 256) {
    float cov = 0.f, tsc = 0.f;
#pragma unroll 8
    for (int s = 0; s < S_; ++s) {
      if (t >= spStart[s] && t < spEnd[s]) {
        tsc += spScale[s];
        cov += 1.0f;
      }
    }
    float np = po[t] * (tsc + (1.0f - cov));
    po[t] = np;
    ltot += np;
  }
  red[tid] = ltot;
  __syncthreads();
  for (int s = 128; s > 0; s >>= 1) {
    if (tid < s) red[tid] += red[tid + s];
    __syncthreads();
  }
  if (tid == 0) sTot = red[0];
  __syncthreads();

  const float inv = 1.0f / sTot;
  for (int t = tid; t < T_; t += 256) po[t] *= inv;
}

// ---------------------------------------------------------------------------
extern "C" void kernel_launch(void* const* d_in, const int* in_sizes, int n_in,
                              void* d_out, int out_size, void* d_ws,
                              size_t ws_size, hipStream_t stream) {
  const float* doc   = (const float*)d_in[0];  // [B,T,D] f32
  const float* qry   = (const float*)d_in[1];  // [B,TQ,D] f32
  const float* mk    = (const float*)d_in[2];  // [B,T] f32
  const int*   spans = (const int*)d_in[3];    // [B,S,2] int32
  const float* sst   = (const float*)d_in[4];  // [B,S] f32
  float*       out   = (float*)d_out;          // [B,T] f32

  float* qpool  = (float*)d_ws;
  float* scores = (float*)((char*)d_ws + (size_t)B_ * D_ * sizeof(float));

  asr_qpool_kernel<<<B_, 256, 0, stream>>>(qry, qpool);
  asr_scores_wmma_kernel<<<B_ * (T_ / 16) / 8, 256, 0, stream>>>(doc, qpool,
                                                                 scores);
  asr_finalize_kernel<<<B_, 256, 0, stream>>>(scores, mk, spans, sst, out);
}